// PatternAnalyzer_2250562863406
// MI455X (gfx1250) — compile-verified
//
#include <hip/hip_runtime.h>
#include <hip/hip_bf16.h>

// ---------------------------------------------------------------------------
// PatternAnalyzer forward for MI455X (gfx1250, wave32, WMMA f32 16x16x4).
// Convs are LDS-staged implicit GEMM (halo zero-padded once at staging; inner
// K-loop is ds_load + wmma only).  mlp1 streams the 128MB weight exactly once
// (memory-bound floor ~5.5us @ 23.3TB/s); activations come from LDS chunks.
// Async-to-LDS (GLOBAL_LOAD_ASYNC_TO_LDS_B128 + s_wait_asynccnt) used for the
// attention-expert feats staging.
// ---------------------------------------------------------------------------

typedef __attribute__((ext_vector_type(2))) float v2f;
typedef __attribute__((ext_vector_type(8))) float v8f;

__device__ __forceinline__ v8f wmma_f32_4(v2f a, v2f b, v8f c) {
  // D = A(16x4) * B(4x16) + C(16x16), fp32
  return __builtin_amdgcn_wmma_f32_16x16x4_f32(false, a, false, b, (short)0, c,
                                               false, false);
}

// ---------------------------------------------------------------------------
// Stage 1: conv1 (1->32, 3x3 SAME) + relu + maxpool2.  K=9: VALU (0.6 GFLOP).
// x:(64,128,128)  cw1:(32,1,3,3)  p1:(64,32,64,64)
// ---------------------------------------------------------------------------
__global__ void __launch_bounds__(256) k_conv1_pool(
    const float* __restrict__ x, const float* __restrict__ cw1,
    const float* __restrict__ cb1, float* __restrict__ p1) {
  int i = blockIdx.x * 256 + threadIdx.x;  // 64*32*64*64 total
  int pw = i & 63, ph = (i >> 6) & 63, c = (i >> 12) & 31, b = i >> 17;
  float w[9];
#pragma unroll
  for (int t = 0; t < 9; ++t) w[t] = cw1[c * 9 + t];
  const float* xb = x + b * 128 * 128;
  float m = -3.4e38f;
#pragma unroll
  for (int sy = 0; sy < 2; ++sy) {
#pragma unroll
    for (int sx = 0; sx < 2; ++sx) {
      int oh = 2 * ph + sy, ow = 2 * pw + sx;
      float acc = 0.f;
#pragma unroll
      for (int ky = 0; ky < 3; ++ky) {
#pragma unroll
        for (int kx = 0; kx < 3; ++kx) {
          int ih = oh + ky - 1, iw = ow + kx - 1;
          float v = (ih >= 0 && ih < 128 && iw >= 0 && iw < 128)
                        ? xb[ih * 128 + iw] : 0.f;
          acc = fmaf(v, w[ky * 3 + kx], acc);
        }
      }
      m = fmaxf(m, acc);
    }
  }
  p1[i] = fmaxf(m + cb1[c], 0.f);  // bias uniform over window, relu monotone
}

// ---------------------------------------------------------------------------
// Stage 2: conv2 (32->64) + relu + pool, LDS-staged implicit GEMM via WMMA.
// Block = (batch b, stripe of 2 pooled rows).  Staged input: 32cin x 6r x 66c
// zero-padded (50.7KB).  koff LUT maps k->cin*396+dy*66+dx.  One wave owns an
// M-tile half: 2 N-tiles x 4 pool positions = 8 accumulators, A reused by all.
// p1:(64,32,64,64) cw2:(64,288) p2:(64,64,32,32)
// ---------------------------------------------------------------------------
__global__ void __launch_bounds__(256) k_conv2_pool_wmma(
    const float* __restrict__ p1, const float* __restrict__ cw2,
    const float* __restrict__ cb2, float* __restrict__ p2) {
  __shared__ float st[32 * 6 * 66];  // 12672 floats
  __shared__ int koff[288];
  int tid = threadIdx.x;
  int b = blockIdx.x >> 4, s = blockIdx.x & 15;  // pooled rows 2s..2s+1
  const float* pb = p1 + b * 32 * 4096;

  for (int k = tid; k < 288; k += 256) {
    int cin = k / 9, r = k - cin * 9, dy = r / 3, dx = r - dy * 3;
    koff[k] = cin * 396 + dy * 66 + dx;
  }
  int gr0 = 4 * s - 1;  // input rows 4s-1..4s+4
  for (int i = tid; i < 12672; i += 256) {
    int cin = i / 396, rem = i - cin * 396;
    int lr = rem / 66, lc = rem - lr * 66;
    int gr = gr0 + lr, gc = lc - 1;
    st[i] = (gr >= 0 && gr < 64 && gc >= 0 && gc < 64)
                ? pb[cin * 4096 + gr * 64 + gc] : 0.f;
  }
  __syncthreads();

  int wave = tid >> 5, lane = tid & 31, half = lane >> 4, l15 = lane & 15;
  int mtile = wave >> 1;          // 0..3 (cout tile)
  int ntbase = (wave & 1) * 2;    // N-tiles {0,1} or {2,3}
  const float* wrow = cw2 + (mtile * 16 + l15) * 288;

  int pixbase[2];
#pragma unroll
  for (int nt = 0; nt < 2; ++nt) {
    int p = (ntbase + nt) * 16 + l15;  // pooled pixel 0..63 in stripe
    pixbase[nt] = ((p >> 5) * 2) * 66 + (p & 31) * 2;
  }

  v8f acc[2][4];
#pragma unroll
  for (int nt = 0; nt < 2; ++nt)
#pragma unroll
    for (int j = 0; j < 4; ++j) acc[nt][j] = v8f{};

#pragma unroll 1
  for (int kk = 0; kk < 288; kk += 4) {
    int kb = kk + half * 2;
    v2f a;
    a.x = wrow[kb];
    a.y = wrow[kb + 1];
    int k0 = koff[kb], k1 = koff[kb + 1];
#pragma unroll
    for (int nt = 0; nt < 2; ++nt) {
#pragma unroll
      for (int j = 0; j < 4; ++j) {  // 2x2 pool position (sy=j>>1, sx=j&1)
        int off = pixbase[nt] + (j >> 1) * 66 + (j & 1);
        v2f bf;
        bf.x = st[k0 + off];
        bf.y = st[k1 + off];
        acc[nt][j] = wmma_f32_4(a, bf, acc[nt][j]);
      }
    }
  }
#pragma unroll
  for (int nt = 0; nt < 2; ++nt) {
    int p = (ntbase + nt) * 16 + l15;
    int pr = 2 * s + (p >> 5), pc = p & 31;
#pragma unroll
    for (int i = 0; i < 8; ++i) {
      int m = mtile * 16 + i + 8 * half;  // cout
      float v = fmaxf(fmaxf(acc[nt][0][i], acc[nt][1][i]),
                      fmaxf(acc[nt][2][i], acc[nt][3][i]));
      p2[(b * 64 + m) * 1024 + pr * 32 + pc] = fmaxf(v + cb2[m], 0.f);
    }
  }
}

// ---------------------------------------------------------------------------
// Stage 3: conv3 (64->128) + relu, LDS-staged implicit GEMM via WMMA.
// Block = (b, 4-row stripe).  Staged input: 64cin x 6r x 34c (52.2KB).
// Wave = M-tile (8 waves = 8 cout tiles); 8 N-tiles per wave, A reused 8x.
// p2:(64,64,32,32) cw3:(128,576) feats:(64,128,32,32)
// ---------------------------------------------------------------------------
__global__ void __launch_bounds__(256) k_conv3_wmma(
    const float* __restrict__ p2, const float* __restrict__ cw3,
    const float* __restrict__ cb3, float* __restrict__ feats) {
  __shared__ float st[64 * 6 * 34];  // 13056 floats
  __shared__ int koff[576];
  int tid = threadIdx.x;
  int b = blockIdx.x >> 3, s = blockIdx.x & 7;  // rows 4s..4s+3
  const float* pb = p2 + b * 64 * 1024;

  for (int k = tid; k < 576; k += 256) {
    int cin = k / 9, r = k - cin * 9, dy = r / 3, dx = r - dy * 3;
    koff[k] = cin * 204 + dy * 34 + dx;
  }
  int gr0 = 4 * s - 1;
  for (int i = tid; i < 13056; i += 256) {
    int cin = i / 204, rem = i - cin * 204;
    int lr = rem / 34, lc = rem - lr * 34;
    int gr = gr0 + lr, gc = lc - 1;
    st[i] = (gr >= 0 && gr < 32 && gc >= 0 && gc < 32)
                ? pb[cin * 1024 + gr * 32 + gc] : 0.f;
  }
  __syncthreads();

  int wave = tid >> 5, lane = tid & 31, half = lane >> 4, l15 = lane & 15;
  const float* wrow = cw3 + (wave * 16 + l15) * 576;  // wave == mtile

  int pixbase[8];
#pragma unroll
  for (int nt = 0; nt < 8; ++nt) {
    int p = nt * 16 + l15;  // pixel 0..127 in stripe
    pixbase[nt] = (p >> 5) * 34 + (p & 31);
  }

  v8f acc[8];
#pragma unroll
  for (int nt = 0; nt < 8; ++nt) acc[nt] = v8f{};

#pragma unroll 1
  for (int kk = 0; kk < 576; kk += 4) {
    int kb = kk + half * 2;
    v2f a;
    a.x = wrow[kb];
    a.y = wrow[kb + 1];
    int k0 = koff[kb], k1 = koff[kb + 1];
#pragma unroll
    for (int nt = 0; nt < 8; ++nt) {
      v2f bf;
      bf.x = st[k0 + pixbase[nt]];
      bf.y = st[k1 + pixbase[nt]];
      acc[nt] = wmma_f32_4(a, bf, acc[nt]);
    }
  }
#pragma unroll
  for (int nt = 0; nt < 8; ++nt) {
#pragma unroll
    for (int i = 0; i < 8; ++i) {
      int m = wave * 16 + i + 8 * half;  // feature channel
      feats[(b * 128 + m) * 1024 + s * 128 + nt * 16 + l15] =
          fmaxf(acc[nt][i] + cb3[m], 0.f);
    }
  }
}

// ---------------------------------------------------------------------------
// Stage 4: fused attention + expert per (b,f).  feats tile staged in LDS via
// GLOBAL_LOAD_ASYNC_TO_LDS_B128 (ASYNCcnt path), softmax in LDS, h1 GEMM via
// WMMA (A from LDS), attn-weighted reduce via ds_add_f32, eo -> z.
// ---------------------------------------------------------------------------
__global__ void __launch_bounds__(256) k_attn_expert(
    const float* __restrict__ feats, const float* __restrict__ attn_w,
    const float* __restrict__ ew1, const float* __restrict__ eb1,
    const float* __restrict__ ew2, const float* __restrict__ eb2,
    float* __restrict__ z) {
  __shared__ float sfe[1024];   // feats[b,f] 32h x 32w
  __shared__ float ssc[32];     // scores
  __shared__ float sattn[32];   // softmax weights
  __shared__ float spool[256];  // pooled

  int tid = threadIdx.x;
  int f = blockIdx.x & 127, b = blockIdx.x >> 7;
  const float* fp = feats + (b * 128 + f) * 1024;

  // Async DMA: each lane moves 16B memory -> LDS (one B128 per lane).
  {
    unsigned lds_dst = (unsigned)(uintptr_t)&sfe[tid * 4];
    unsigned long long gsrc = (unsigned long long)(uintptr_t)(fp + tid * 4);
    asm volatile("global_load_async_to_lds_b128 %0, %1, off"
                 :: "v"(lds_dst), "v"(gsrc) : "memory");
  }
  spool[tid & 255] = 0.f;
  asm volatile("s_wait_asynccnt 0" ::: "memory");
  __syncthreads();

  if (tid < 32) {
    float s = 0.f;
#pragma unroll
    for (int w = 0; w < 32; ++w)
      s = fmaf(sfe[tid * 32 + w], attn_w[f * 32 + w], s);
    ssc[tid] = s;
  }
  __syncthreads();
  if (tid < 32) {
    float mx = ssc[0];
    for (int h2 = 1; h2 < 32; ++h2) mx = fmaxf(mx, ssc[h2]);
    float den = 0.f;
    for (int h2 = 0; h2 < 32; ++h2) den += __expf(ssc[h2] - mx);
    sattn[tid] = __expf(ssc[tid] - mx) / den;
  }
  __syncthreads();

  int wave = tid >> 5, lane = tid & 31, half = lane >> 4, l15 = lane & 15;
  const float* ewf = ew1 + f * 32 * 256;
#pragma unroll 1
  for (int q = 0; q < 4; ++q) {
    int t = wave * 4 + q;  // 32 tiles: 2 mtiles x 16 ntiles
    int mt = t & 1, nt = t >> 1;
    int ncol = nt * 16 + l15;
    v8f acc = v8f{};
#pragma unroll
    for (int kk = 0; kk < 32; kk += 4) {
      int kb = kk + half * 2;
      v2f a;
      a.x = sfe[(mt * 16 + l15) * 32 + kb];
      a.y = sfe[(mt * 16 + l15) * 32 + kb + 1];
      v2f bf;
      bf.x = ewf[kb * 256 + ncol];
      bf.y = ewf[(kb + 1) * 256 + ncol];
      acc = wmma_f32_4(a, bf, acc);
    }
    float bias = eb1[f * 256 + ncol];
#pragma unroll
    for (int i = 0; i < 8; ++i) {
      int h2 = mt * 16 + i + 8 * half;  // row = h
      float v = fmaxf(acc[i] + bias, 0.f) * sattn[h2];
      atomicAdd(&spool[ncol], v);  // ds_add_f32: sum over h
    }
  }
  __syncthreads();

  if (tid < 64) {
    const float* w2 = ew2 + f * 256 * 64;
    float s = eb2[f * 64 + tid];
#pragma unroll 4
    for (int k = 0; k < 256; ++k) s = fmaf(spool[k], w2[k * 64 + tid], s);
    z[b * 8192 + f * 64 + tid] = s;
  }
}

// ---------------------------------------------------------------------------
// Stage 5: mlp1  z(64x8192) @ mw1(8192x4096) + mb1, relu -> zh(64x4096).
// Memory-bound: mw1=128MB streamed exactly once (wave owns one 16-col N-tile,
// all 4 M-tiles).  Activations staged in LDS in 128-K chunks (coalesced).
// ---------------------------------------------------------------------------
__global__ void __launch_bounds__(256) k_mlp1_wmma(
    const float* __restrict__ z, const float* __restrict__ mw1,
    const float* __restrict__ mb1, float* __restrict__ zh) {
  __shared__ float sz[64 * 128];  // 32KB chunk of z
  int tid = threadIdx.x;
  int wave = tid >> 5, lane = tid & 31, half = lane >> 4, l15 = lane & 15;
  int n = (blockIdx.x * 8 + wave) * 16 + l15;  // ntile 0..255

  v8f acc[4] = {v8f{}, v8f{}, v8f{}, v8f{}};
#pragma unroll 1
  for (int kc = 0; kc < 8192; kc += 128) {
    __syncthreads();
    for (int i = tid; i < 8192; i += 256)
      sz[i] = z[(i >> 7) * 8192 + kc + (i & 127)];
    __syncthreads();
#pragma unroll 1
    for (int kk = 0; kk < 128; kk += 4) {
      int kb = kk + half * 2;
      __builtin_prefetch(&mw1[(size_t)(kc + kb + 128) * 4096 + n], 0, 1);
      v2f bf;
      bf.x = mw1[(size_t)(kc + kb) * 4096 + n];
      bf.y = mw1[(size_t)(kc + kb + 1) * 4096 + n];
#pragma unroll
      for (int mt = 0; mt < 4; ++mt) {
        v2f a;
        a.x = sz[(mt * 16 + l15) * 128 + kb];
        a.y = sz[(mt * 16 + l15) * 128 + kb + 1];
        acc[mt] = wmma_f32_4(a, bf, acc[mt]);
      }
    }
  }
  float bias = mb1[n];
#pragma unroll
  for (int mt = 0; mt < 4; ++mt) {
#pragma unroll
    for (int i = 0; i < 8; ++i) {
      int m = mt * 16 + i + 8 * half;
      zh[m * 4096 + n] = fmaxf(acc[mt][i] + bias, 0.f);
    }
  }
}

// ---------------------------------------------------------------------------
// Stage 6: mlp2  zh(64x4096) @ mw2(4096x256) + mb2 -> out(64x256).
// ---------------------------------------------------------------------------
__global__ void __launch_bounds__(256) k_mlp2_wmma(
    const float* __restrict__ zh, const float* __restrict__ mw2,
    const float* __restrict__ mb2, float* __restrict__ out) {
  __shared__ float sz[64 * 128];  // 32KB chunk of zh
  int tid = threadIdx.x;
  int wave = tid >> 5, lane = tid & 31, half = lane >> 4, l15 = lane & 15;
  int n = (blockIdx.x * 8 + wave) * 16 + l15;  // ntile 0..15

  v8f acc[4] = {v8f{}, v8f{}, v8f{}, v8f{}};
#pragma unroll 1
  for (int kc = 0; kc < 4096; kc += 128) {
    __syncthreads();
    for (int i = tid; i < 8192; i += 256)
      sz[i] = zh[(i >> 7) * 4096 + kc + (i & 127)];
    __syncthreads();
#pragma unroll 1
    for (int kk = 0; kk < 128; kk += 4) {
      int kb = kk + half * 2;
      v2f bf;
      bf.x = mw2[(kc + kb) * 256 + n];
      bf.y = mw2[(kc + kb + 1) * 256 + n];
#pragma unroll
      for (int mt = 0; mt < 4; ++mt) {
        v2f a;
        a.x = sz[(mt * 16 + l15) * 128 + kb];
        a.y = sz[(mt * 16 + l15) * 128 + kb + 1];
        acc[mt] = wmma_f32_4(a, bf, acc[mt]);
      }
    }
  }
  float bias = mb2[n];
#pragma unroll
  for (int mt = 0; mt < 4; ++mt) {
#pragma unroll
    for (int i = 0; i < 8; ++i) {
      int m = mt * 16 + i + 8 * half;
      out[m * 256 + n] = acc[mt][i] + bias;
    }
  }
}

// ---------------------------------------------------------------------------
extern "C" void kernel_launch(void* const* d_in, const int* in_sizes, int n_in,
                              void* d_out, int out_size, void* d_ws,
                              size_t ws_size, hipStream_t stream) {
  const float* x      = (const float*)d_in[0];
  const float* cw1    = (const float*)d_in[1];
  const float* cb1    = (const float*)d_in[2];
  const float* cw2    = (const float*)d_in[3];
  const float* cb2    = (const float*)d_in[4];
  const float* cw3    = (const float*)d_in[5];
  const float* cb3    = (const float*)d_in[6];
  const float* attn_w = (const float*)d_in[7];
  const float* ew1    = (const float*)d_in[8];
  const float* eb1    = (const float*)d_in[9];
  const float* ew2    = (const float*)d_in[10];
  const float* eb2    = (const float*)d_in[11];
  const float* mw1    = (const float*)d_in[12];
  const float* mb1    = (const float*)d_in[13];
  const float* mw2    = (const float*)d_in[14];
  const float* mb2    = (const float*)d_in[15];

  float* ws    = (float*)d_ws;
  float* p1    = ws;               // 64*32*64*64 = 8388608 f
  float* p2    = p1 + 8388608;     // 64*64*32*32 = 4194304 f
  float* feats = p2 + 4194304;     // 64*128*32*32 = 8388608 f
  float* zbuf  = feats + 8388608;  // 64*8192 = 524288 f
  float* zh    = zbuf + 524288;    // 64*4096 = 262144 f
  float* out   = (float*)d_out;    // 64*256

  k_conv1_pool<<<32768, 256, 0, stream>>>(x, cw1, cb1, p1);
  k_conv2_pool_wmma<<<1024, 256, 0, stream>>>(p1, cw2, cb2, p2);
  k_conv3_wmma<<<512, 256, 0, stream>>>(p2, cw3, cb3, feats);
  k_attn_expert<<<8192, 256, 0, stream>>>(feats, attn_w, ew1, eb1, ew2, eb2,
                                          zbuf);
  k_mlp1_wmma<<<32, 256, 0, stream>>>(zbuf, mw1, mb1, zh);
  k_mlp2_wmma<<<2, 256, 0, stream>>>(zh, mw2, mb2, out);
}